// Seq2Seq_5755256176945
// MI455X (gfx1250) — compile-verified
//
#include <hip/hip_runtime.h>
#include <hip/hip_bf16.h>
#include <math.h>

// ---------------------------------------------------------------------------
// Seq2Seq GRU (encoder 128 steps, decoder 64 steps, vocab projection)
// fp32 path using V_WMMA_F32_16X16X4_F32 (wave32, gfx1250). All matrix work
// (big GEMMs *and* the recurrent step) is LDS-staged and double-buffered,
// fed by GLOBAL_LOAD_ASYNC_TO_LDS_B128 (ASYNCcnt) with pipelined
// s_wait_asynccnt thresholds.
//
// Fragment layout assumptions (CDNA5 ISA 7.12.2, 32-bit matrices):
//   A (16x4, MxK): lane L<16  -> {X[m0+L][k0+0], X[m0+L][k0+1]}
//                  lane L>=16 -> {X[m0+L-16][k0+2], X[m0+L-16][k0+3]}
//   B (4x16, KxN): mirrored from W rows (W is (N,K) row-major)
//   C/D (16x16):   lane L<16  : VGPR v -> C[m0+v][n0+L]
//                  lane L>=16 : VGPR v -> C[m0+8+v][n0+L-16]
// ---------------------------------------------------------------------------

typedef __attribute__((ext_vector_type(2))) float v2f;
typedef __attribute__((ext_vector_type(8))) float v8f;

__device__ __forceinline__ v8f wmma4_f32(v2f a, v2f b, v8f c) {
  return __builtin_amdgcn_wmma_f32_16x16x4_f32(false, a, false, b, (short)0, c,
                                               false, false);
}

// Async global->LDS copy, 16 bytes per lane (512 B per wave instruction).
__device__ __forceinline__ void s2s_async_copy16(unsigned int lds_off,
                                                 const float* gptr) {
  asm volatile("global_load_async_to_lds_b128 %0, %1, off"
               :
               : "v"(lds_off), "v"(gptr)
               : "memory");
}

__device__ __forceinline__ unsigned int s2s_lds_off(const void* p) {
  // Generic pointers to LDS carry the LDS byte offset in their low 32 bits
  // (ISA 10.2 aperture mapping).
  return (unsigned int)(unsigned long long)p;
}

__device__ __forceinline__ void s2s_wait_async0() {
#if __has_builtin(__builtin_amdgcn_s_wait_asynccnt)
  __builtin_amdgcn_s_wait_asynccnt((short)0);
#else
  asm volatile("s_wait_asynccnt 0x0" ::: "memory");
#endif
}

__device__ __forceinline__ void s2s_wait_async5() {
#if __has_builtin(__builtin_amdgcn_s_wait_asynccnt)
  __builtin_amdgcn_s_wait_asynccnt((short)5);
#else
  asm volatile("s_wait_asynccnt 0x5" ::: "memory");
#endif
}

__device__ __forceinline__ void s2s_wait_async6() {
#if __has_builtin(__builtin_amdgcn_s_wait_asynccnt)
  __builtin_amdgcn_s_wait_asynccnt((short)6);
#else
  asm volatile("s_wait_asynccnt 0x6" ::: "memory");
#endif
}

#define LDSTRIDE 36  // 32 floats + 4 pad: 144 B rows, 16B-aligned, no bank dup

// ---------------------------------------------------------------------------
// Embedding gathers (build time-major activation matrices, row m = t*B + b)
// ---------------------------------------------------------------------------
__global__ void s2s_gather_enc(const int* __restrict__ x,
                               const float* __restrict__ emb,
                               float* __restrict__ XE) {
  int m = blockIdx.x;          // 0..8191
  int t = m >> 6, b = m & 63;  // B = 64
  int tok = x[b * 128 + t];    // x: (B, 128)
  const float4* src = (const float4*)(emb + (size_t)tok * 512);
  float4* dst = (float4*)(XE + (size_t)m * 512);
  dst[threadIdx.x] = src[threadIdx.x];  // 128 threads x 16 B
}

__global__ void s2s_gather_dec(const int* __restrict__ labels,
                               const int* __restrict__ bosp,
                               const float* __restrict__ emb,
                               float* __restrict__ XW) {
  int m = blockIdx.x;  // 0..4095
  int t = m >> 6, b = m & 63;
  int tok = (t == 0) ? bosp[0] : labels[b * 64 + (t - 1)];
  const float4* src = (const float4*)(emb + (size_t)tok * 512);
  float4* dst = (float4*)(XW + (size_t)m * 512);
  dst[threadIdx.x] = src[threadIdx.x];
}

// ---------------------------------------------------------------------------
// Init helpers
// ---------------------------------------------------------------------------
__global__ void s2s_fill_zero(float* __restrict__ p, int n) {
  int i = blockIdx.x * blockDim.x + threadIdx.x;
  if (i < n) p[i] = 0.0f;
}

__global__ void s2s_bcast_row(const float* __restrict__ row,
                              float* __restrict__ dst, int H, int n) {
  int i = blockIdx.x * blockDim.x + threadIdx.x;
  if (i < n) dst[i] = row[i % H];
}

// ---------------------------------------------------------------------------
// Tiled GEMM: Y = X @ W^T (+bias). X:(M,K) row-major (ld=ldx), W:(N,K)
// row-major (ld=ldw). Block (256 threads, 8 waves) computes 64x128 of Y.
// K-chunks of 32 staged in double-buffered LDS via async b128 copies.
// out_btv: 0 -> Y[m*ldy + n]; 1 -> logits scatter Y[((m&63)*64 + (m>>6))*ldy+n]
// Requires M%64==0, N%128==0, K%32==0; grid = (M/64)*(N/128) blocks.
// ---------------------------------------------------------------------------
__device__ __forceinline__ void s2s_fill_chunk(const float* __restrict__ Xg,
                                               int ldx,
                                               const float* __restrict__ Wg,
                                               int ldw, int k0,
                                               float* __restrict__ sa,
                                               float* __restrict__ sb,
                                               int tid) {
  // A tile: 64 rows x 32 floats = 512 float4 -> 2 per thread
#pragma unroll
  for (int q = 0; q < 2; ++q) {
    int e = tid + q * 256;
    int row = e >> 3, c4 = e & 7;
    s2s_async_copy16(s2s_lds_off(&sa[row * LDSTRIDE + c4 * 4]),
                     Xg + (size_t)row * ldx + k0 + c4 * 4);
  }
  // B tile: 128 rows x 32 floats = 1024 float4 -> 4 per thread
#pragma unroll
  for (int q = 0; q < 4; ++q) {
    int e = tid + q * 256;
    int row = e >> 3, c4 = e & 7;
    s2s_async_copy16(s2s_lds_off(&sb[row * LDSTRIDE + c4 * 4]),
                     Wg + (size_t)row * ldw + k0 + c4 * 4);
  }
}

__global__ __launch_bounds__(256) void s2s_gemm_tiled(
    const float* __restrict__ X, int ldx, const float* __restrict__ W, int ldw,
    const float* __restrict__ bias, float* __restrict__ Y, int ldy, int Mblocks,
    int K, int out_btv) {
  __shared__ float sA[2][64 * LDSTRIDE];
  __shared__ float sB[2][128 * LDSTRIDE];

  int tid = threadIdx.x;
  int wid = tid >> 5;
  int lane = tid & 31;
  int half = lane >> 4;
  int r = lane & 15;
  int wm = wid & 3;   // 4 M strips of 16
  int wn = wid >> 2;  // 2 N strips of 64

  int mb = blockIdx.x % Mblocks;
  int nb = blockIdx.x / Mblocks;
  int m0 = mb * 64;
  int n0 = nb * 128;

  const float* Xg = X + (size_t)m0 * ldx;
  const float* Wg = W + (size_t)n0 * ldw;

  v8f acc0 = {}, acc1 = {}, acc2 = {}, acc3 = {};

  int nk = K / 32;
  s2s_fill_chunk(Xg, ldx, Wg, ldw, 0, sA[0], sB[0], tid);

  for (int i = 0; i < nk; ++i) {
    if (i + 1 < nk) {
      s2s_fill_chunk(Xg, ldx, Wg, ldw, (i + 1) * 32, sA[(i + 1) & 1],
                     sB[(i + 1) & 1], tid);
      s2s_wait_async6();  // chunk i complete; chunk i+1 still in flight
    } else {
      s2s_wait_async0();
    }
    __syncthreads();

    const float* As = &sA[i & 1][(wm * 16 + r) * LDSTRIDE + 2 * half];
    const float* B0 = &sB[i & 1][(wn * 64 + 0 + r) * LDSTRIDE + 2 * half];
    const float* B1 = &sB[i & 1][(wn * 64 + 16 + r) * LDSTRIDE + 2 * half];
    const float* B2 = &sB[i & 1][(wn * 64 + 32 + r) * LDSTRIDE + 2 * half];
    const float* B3 = &sB[i & 1][(wn * 64 + 48 + r) * LDSTRIDE + 2 * half];

#pragma unroll
    for (int k = 0; k < 32; k += 4) {
      v2f a = {As[k], As[k + 1]};
      v2f b0 = {B0[k], B0[k + 1]};
      v2f b1 = {B1[k], B1[k + 1]};
      v2f b2 = {B2[k], B2[k + 1]};
      v2f b3 = {B3[k], B3[k + 1]};
      acc0 = wmma4_f32(a, b0, acc0);
      acc1 = wmma4_f32(a, b1, acc1);
      acc2 = wmma4_f32(a, b2, acc2);
      acc3 = wmma4_f32(a, b3, acc3);
    }
    __syncthreads();
  }

  int c0 = n0 + wn * 64 + 0 + r;
  int c1 = c0 + 16, c2 = c0 + 32, c3 = c0 + 48;
  float bi0 = bias ? bias[c0] : 0.0f;
  float bi1 = bias ? bias[c1] : 0.0f;
  float bi2 = bias ? bias[c2] : 0.0f;
  float bi3 = bias ? bias[c3] : 0.0f;

#pragma unroll
  for (int v = 0; v < 8; ++v) {
    int m = m0 + wm * 16 + 8 * half + v;
    size_t base;
    if (out_btv) {
      int t = m >> 6, b = m & 63;  // row m = t*64 + b ; out is (b, t, v)
      base = ((size_t)b * 64 + t) * (size_t)ldy;
    } else {
      base = (size_t)m * (size_t)ldy;
    }
    Y[base + c0] = acc0[v] + bi0;
    Y[base + c1] = acc1[v] + bi1;
    Y[base + c2] = acc2[v] + bi2;
    Y[base + c3] = acc3[v] + bi3;
  }
}

// ---------------------------------------------------------------------------
// One GRU timestep (B=64 fixed), LDS-staged like the big GEMMs.
// Block (8 waves) owns a 32-column hidden slice: all 64 batch rows x 32 cols
// x 3 gates. Per K-chunk of 32 it stages A = h tile (64x32) and B = the 96
// Whh rows {j, H+j, 2H+j : j in slice}, double buffered (46 KB LDS).
// Wave (wb, wj) = 16 batch rows x 16 cols, 3 accumulators (r/z/n) sharing
// one A fragment. Gate math fused in the epilogue.
// Grid: H/32 = 32 blocks x 256 threads (256 waves).
// ---------------------------------------------------------------------------
__device__ __forceinline__ void s2s_gru_fill(const float* __restrict__ h_in,
                                             const float* __restrict__ Whh,
                                             int H, int j0, int k0,
                                             float* __restrict__ sa,
                                             float* __restrict__ sb, int tid) {
  // A tile: 64 rows x 32 floats = 512 float4 -> 2 per thread
#pragma unroll
  for (int q = 0; q < 2; ++q) {
    int e = tid + q * 256;
    int row = e >> 3, c4 = e & 7;
    s2s_async_copy16(s2s_lds_off(&sa[row * LDSTRIDE + c4 * 4]),
                     h_in + (size_t)row * H + k0 + c4 * 4);
  }
  // B tile: 96 rows (3 gates x 32 cols) x 32 floats = 768 float4 -> 3/thread
#pragma unroll
  for (int q = 0; q < 3; ++q) {
    int e = tid + q * 256;
    int row = e >> 3, c4 = e & 7;  // row 0..95
    int gate = row >> 5, jj = row & 31;
    s2s_async_copy16(s2s_lds_off(&sb[row * LDSTRIDE + c4 * 4]),
                     Whh + (size_t)(gate * H + j0 + jj) * H + k0 + c4 * 4);
  }
}

__global__ __launch_bounds__(256) void s2s_gru_step(
    const float* __restrict__ h_in, const float* __restrict__ Whh,
    const float* __restrict__ bhh, const float* __restrict__ gi,
    const float* __restrict__ gctx, float* __restrict__ h_out,
    float* __restrict__ states_out, int H) {
  __shared__ float sA[2][64 * LDSTRIDE];
  __shared__ float sB[2][96 * LDSTRIDE];

  int tid = threadIdx.x;
  int wid = tid >> 5;
  int lane = tid & 31;
  int half = lane >> 4;
  int r = lane & 15;
  int wj = wid & 1;   // 2 column subtiles of 16
  int wb = wid >> 1;  // 4 batch-row tiles of 16
  int j0 = blockIdx.x * 32;

  v8f aR = {}, aZ = {}, aN = {};

  int nk = H / 32;
  s2s_gru_fill(h_in, Whh, H, j0, 0, sA[0], sB[0], tid);

  for (int i = 0; i < nk; ++i) {
    if (i + 1 < nk) {
      s2s_gru_fill(h_in, Whh, H, j0, (i + 1) * 32, sA[(i + 1) & 1],
                   sB[(i + 1) & 1], tid);
      s2s_wait_async5();  // chunk i done; chunk i+1 (5 ops/wave) in flight
    } else {
      s2s_wait_async0();
    }
    __syncthreads();

    const float* As = &sA[i & 1][(wb * 16 + r) * LDSTRIDE + 2 * half];
    const float* BR = &sB[i & 1][(0 + wj * 16 + r) * LDSTRIDE + 2 * half];
    const float* BZ = &sB[i & 1][(32 + wj * 16 + r) * LDSTRIDE + 2 * half];
    const float* BN = &sB[i & 1][(64 + wj * 16 + r) * LDSTRIDE + 2 * half];

#pragma unroll
    for (int k = 0; k < 32; k += 4) {
      v2f a = {As[k], As[k + 1]};
      v2f bR = {BR[k], BR[k + 1]};
      v2f bZ = {BZ[k], BZ[k + 1]};
      v2f bN = {BN[k], BN[k + 1]};
      aR = wmma4_f32(a, bR, aR);
      aZ = wmma4_f32(a, bZ, aZ);
      aN = wmma4_f32(a, bN, aN);
    }
    __syncthreads();
  }

  int j = j0 + wj * 16 + r;
  int b0 = wb * 16;
  float bhr = bhh[j], bhz = bhh[H + j], bhn = bhh[2 * H + j];
#pragma unroll
  for (int v = 0; v < 8; ++v) {
    int b = b0 + 8 * half + v;
    size_t g = (size_t)b * 3 * H;
    float gir = gi[g + j];
    float giz = gi[g + H + j];
    float gin = gi[g + 2 * H + j];
    if (gctx) {
      gir += gctx[g + j];
      giz += gctx[g + H + j];
      gin += gctx[g + 2 * H + j];
    }
    float ghr = aR[v] + bhr;
    float ghz = aZ[v] + bhz;
    float ghn = aN[v] + bhn;
    float rg = 1.0f / (1.0f + expf(-(gir + ghr)));
    float zg = 1.0f / (1.0f + expf(-(giz + ghz)));
    float ng = tanhf(gin + rg * ghn);
    float hp = h_in[(size_t)b * H + j];
    float hn = (1.0f - zg) * ng + zg * hp;
    h_out[(size_t)b * H + j] = hn;
    if (states_out) states_out[(size_t)b * H + j] = hn;
  }
}

// ---------------------------------------------------------------------------
// Host orchestration
// ---------------------------------------------------------------------------
extern "C" void kernel_launch(void* const* d_in, const int* in_sizes, int n_in,
                              void* d_out, int out_size, void* d_ws,
                              size_t ws_size, hipStream_t stream) {
  (void)in_sizes; (void)n_in; (void)out_size; (void)ws_size;

  const int* x = (const int*)d_in[0];        // (64,128)
  const int* labels = (const int*)d_in[1];   // (64,64)
  const int* bos = (const int*)d_in[2];      // scalar
  const float* enc_emb = (const float*)d_in[3];   // (32000,512)
  const float* enc_Wih = (const float*)d_in[4];   // (3072,512)
  const float* enc_Whh = (const float*)d_in[5];   // (3072,1024)
  const float* enc_bih = (const float*)d_in[6];   // (3072)
  const float* enc_bhh = (const float*)d_in[7];   // (3072)
  const float* dec_emb = (const float*)d_in[8];   // (32000,512)
  const float* dec_Wih = (const float*)d_in[9];   // (3072,1536)
  const float* dec_Whh = (const float*)d_in[10];  // (3072,1024)
  const float* dec_bih = (const float*)d_in[11];  // (3072)
  const float* dec_bhh = (const float*)d_in[12];  // (3072)
  const float* dec_init = (const float*)d_in[13]; // (1,1024)
  const float* lin_W = (const float*)d_in[14];    // (32000,1024)
  const float* lin_b = (const float*)d_in[15];    // (32000)
  float* out = (float*)d_out;                     // (64,64,32000)

  const int B = 64, TENC = 128, TDEC = 64, H = 1024, DE = 512;
  const int G3 = 3 * H;  // 3072

  // Workspace layout (floats)
  float* ws = (float*)d_ws;
  float* Gx = ws;                              // (8192, 3072)
  float* XE = Gx + (size_t)TENC * B * G3;      // (8192, 512)
  float* XW = XE;                              // reused after Gx GEMM
  float* Gw = XE + (size_t)TENC * B * DE;      // (4096, 3072)
  float* Gc = Gw + (size_t)TDEC * B * G3;      // (64, 3072)
  float* Sts = Gc + (size_t)B * G3;            // (64, 64, 1024)
  float* hA = Sts + (size_t)TDEC * B * H;      // (B,H) x4 ping-pong
  float* hB = hA + (size_t)B * H;
  float* hC = hB + (size_t)B * H;
  float* hD = hC + (size_t)B * H;

  // ---- Encoder ----
  s2s_gather_enc<<<dim3(TENC * B), dim3(128), 0, stream>>>(x, enc_emb, XE);

  // Gx = XE @ enc_Wih^T + enc_bih : M=8192, N=3072, K=512
  s2s_gemm_tiled<<<dim3((TENC * B / 64) * (G3 / 128)), dim3(256), 0, stream>>>(
      XE, DE, enc_Wih, DE, enc_bih, Gx, G3, TENC * B / 64, DE, 0);

  s2s_fill_zero<<<dim3((B * H + 255) / 256), dim3(256), 0, stream>>>(hA, B * H);

  for (int t = 0; t < TENC; ++t) {
    const float* hin = (t & 1) ? hB : hA;
    float* hout = (t & 1) ? hA : hB;
    s2s_gru_step<<<dim3(H / 32), dim3(256), 0, stream>>>(
        hin, enc_Whh, enc_bhh, Gx + (size_t)t * B * G3, nullptr, hout, nullptr,
        H);
  }
  const float* e_final = hA;  // TENC even -> final state back in hA

  // ---- Decoder precompute ----
  s2s_gather_dec<<<dim3(TDEC * B), dim3(128), 0, stream>>>(labels, bos, dec_emb,
                                                           XW);
  // Gw = XW @ dec_Wih[:, :512]^T + dec_bih : M=4096, N=3072, K=512, ldw=1536
  s2s_gemm_tiled<<<dim3((TDEC * B / 64) * (G3 / 128)), dim3(256), 0, stream>>>(
      XW, DE, dec_Wih, DE + H, dec_bih, Gw, G3, TDEC * B / 64, DE, 0);
  // Gc = e_final @ dec_Wih[:, 512:]^T : M=64, N=3072, K=1024
  s2s_gemm_tiled<<<dim3(1 * (G3 / 128)), dim3(256), 0, stream>>>(
      e_final, H, dec_Wih + DE, DE + H, nullptr, Gc, G3, 1, H, 0);

  s2s_bcast_row<<<dim3((B * H + 255) / 256), dim3(256), 0, stream>>>(
      dec_init, hC, H, B * H);

  for (int t = 0; t < TDEC; ++t) {
    const float* hin = (t & 1) ? hD : hC;
    float* hout = (t & 1) ? hC : hD;
    s2s_gru_step<<<dim3(H / 32), dim3(256), 0, stream>>>(
        hin, dec_Whh, dec_bhh, Gw + (size_t)t * B * G3, Gc, hout,
        Sts + (size_t)t * B * H, H);
  }

  // ---- Vocab projection: (4096 x 1024) @ (1024 x 32000) -> btv scatter ----
  s2s_gemm_tiled<<<dim3((TDEC * B / 64) * (32000 / 128)), dim3(256), 0,
                   stream>>>(Sts, H, lin_W, H, lin_b, out, 32000, TDEC * B / 64,
                             H, 1);
}